// SparseResBlock_14723147891281
// MI455X (gfx1250) — compile-verified
//
#include <hip/hip_runtime.h>

// SparseResBlock on gfx1250: GN folded into weights, gather-GEMMs on
// V_WMMA_F32_16X16X4_F32 (fp32 matrix path, matches reference precision).
// Each wave computes a 32-voxel x 64-output tile: two A fragments share every
// B fragment (2 WMMAs per ds_load_b64).

typedef float v2f __attribute__((ext_vector_type(2)));
typedef float v8f __attribute__((ext_vector_type(8)));

#define KTAPS 27
#define PB 512  // reduction partial blocks

__device__ __forceinline__ float silu_f(float v) {
    return v / (1.0f + expf(-v));
}

// ---------------- GroupNorm stats: stage 1 (per-block partials) --------------
template<int CIN>
__global__ __launch_bounds__(256) void gn_partials(const float* __restrict__ x,
                                                   int N,
                                                   float* __restrict__ part) {
    constexpr int CPG = CIN / 8;  // channels per group (G=8)
    float s[8], q[8];
#pragma unroll
    for (int g = 0; g < 8; ++g) { s[g] = 0.f; q[g] = 0.f; }
    for (int i = blockIdx.x * 256 + threadIdx.x; i < N; i += gridDim.x * 256) {
        const float4* p = (const float4*)(x + (size_t)i * CIN);
#pragma unroll
        for (int j = 0; j < CIN / 4; ++j) {
            float4 v = p[j];
            int g = (j * 4) / CPG;
            s[g] += v.x + v.y + v.z + v.w;
            q[g] += v.x * v.x + v.y * v.y + v.z * v.z + v.w * v.w;
        }
    }
    __shared__ float red[256 * 16];
    int tid = threadIdx.x;
#pragma unroll
    for (int g = 0; g < 8; ++g) { red[tid * 16 + g] = s[g]; red[tid * 16 + 8 + g] = q[g]; }
    __syncthreads();
    for (int st = 128; st > 0; st >>= 1) {
        if (tid < st) {
#pragma unroll
            for (int j = 0; j < 16; ++j) red[tid * 16 + j] += red[(tid + st) * 16 + j];
        }
        __syncthreads();
    }
    if (tid < 16) part[(size_t)blockIdx.x * 16 + tid] = red[tid];
}

// ------------- GroupNorm stats: stage 2 (fixed-order final + fold) ----------
// a[c] = gamma[c]*rsqrt(var+eps); b[c] = beta[c] - mean*a[c]
// Cb[n] = sum_k sum_c b[c] * W[k][c][n]  (constant term from folding GN)
template<int CIN>
__global__ __launch_bounds__(256) void gn_finalize(const float* __restrict__ part,
                                                   float invCnt,
                                                   const float* __restrict__ gamma,
                                                   const float* __restrict__ beta,
                                                   const float* __restrict__ W,
                                                   float* __restrict__ ab,
                                                   float* __restrict__ Cb) {
    __shared__ float tot[16];
    __shared__ float bsh[CIN];
    int tid = threadIdx.x;
    if (tid < 16) {
        float s = 0.f;
        for (int p = 0; p < PB; ++p) s += part[p * 16 + tid];  // fixed order
        tot[tid] = s;
    }
    __syncthreads();
    if (tid < CIN) {
        int g = tid / (CIN / 8);
        float mean = tot[g] * invCnt;
        float var  = tot[8 + g] * invCnt - mean * mean;
        float a = gamma[tid] * rsqrtf(var + 1e-5f);
        float b = beta[tid] - mean * a;
        ab[tid] = a; ab[CIN + tid] = b; bsh[tid] = b;
    }
    __syncthreads();
    if (tid < 64) {
        float acc = 0.f;
        for (int k = 0; k < KTAPS; ++k) {
            const float* wk = W + (size_t)k * CIN * 64 + tid;
            for (int c = 0; c < CIN; ++c) acc += bsh[c] * wk[c * 64];
        }
        Cb[tid] = acc;
    }
}

// ---- Pre-scale weights by a[] and repack into WMMA-B pair layout -----------
// Wp[k] is (CIN/2) x 64 of float2 {W[2r][n], W[2r+1][n]} so one ds_load_b64
// per lane yields the 4x16 B fragment of V_WMMA_F32_16X16X4_F32.
template<int CIN>
__global__ __launch_bounds__(256) void prep_w(const float* __restrict__ W,
                                              const float* __restrict__ ab,
                                              float* __restrict__ Wp) {
    __shared__ float ash[CIN];
    int k = blockIdx.x, tid = threadIdx.x;
    if (tid < CIN) ash[tid] = ab ? ab[tid] : 1.0f;
    __syncthreads();
    const float* wsrc = W + (size_t)k * CIN * 64;
    float2* wd = (float2*)(Wp + (size_t)k * CIN * 64);
    for (int e = tid; e < (CIN / 2) * 64; e += 256) {
        int r = e >> 6, n = e & 63;
        float2 o;
        o.x = ash[2 * r]     * wsrc[(2 * r) * 64 + n];
        o.y = ash[2 * r + 1] * wsrc[(2 * r + 1) * 64 + n];
        wd[e] = o;
    }
}

// ---------------- Gather sparse conv via fp32 WMMA --------------------------
// One wave = 32 voxels x 64 outputs (8 accumulators of v8f).
// K loop: 27 taps x (CIN/4) chunks of K=4; B staged in LDS per tap and shared
// by the two 16-voxel halves of the wave's tile.
template<int CIN, bool SKIP>
__global__ __launch_bounds__(256) void sparse_conv_wmma(
        const float* __restrict__ act,      // [N, CIN] activation (gathered)
        const int*   __restrict__ nbr,      // [N, 27]
        const float* __restrict__ Wp,       // [27, CIN/2, 64] float2-packed
        const float* __restrict__ Cb,       // [64] folded-GN constant
        const float* __restrict__ skipAct,  // [N, 32] (=x, SKIP only)
        const float* __restrict__ WpSkip,   // [16, 64] float2-packed (SKIP only)
        const float* __restrict__ bskip,    // [64] (SKIP only)
        float* __restrict__ out,            // [N, 64]
        int N, int nPairs) {
    __shared__ float wlds[CIN * 64];
    const int tid  = threadIdx.x;
    const int wave = tid >> 5;
    const int lane = tid & 31;
    const int lo   = lane & 15;
    const int hi   = lane >> 4;  // 0: K rows 4c,4c+1 ; 1: K rows 4c+2,4c+3

    int pair = blockIdx.x * 8 + wave;
    if (pair >= nPairs) pair = nPairs - 1;  // duplicate last tile; barrier-safe
    const int m0 = pair << 5;               // 32 voxels per wave
    int vm0 = m0 + lo;       if (vm0 >= N) vm0 = N - 1;
    int vm1 = m0 + 16 + lo;  if (vm1 >= N) vm1 = N - 1;

    v8f accA[4] = {};
    v8f accB[4] = {};

    for (int k = 0; k < KTAPS; ++k) {
        // stage this tap's pre-packed weights into LDS (uniform loop)
        const float4* src = (const float4*)(Wp + (size_t)k * CIN * 64);
#pragma unroll
        for (int i = tid; i < CIN * 16; i += 256) ((float4*)wlds)[i] = src[i];
        __syncthreads();

        const int r0 = nbr[(size_t)vm0 * KTAPS + k];
        const int r1 = nbr[(size_t)vm1 * KTAPS + k];
        const float* pa0 = act + (size_t)r0 * CIN + 2 * hi;
        const float* pa1 = act + (size_t)r1 * CIN + 2 * hi;
#pragma unroll
        for (int c = 0; c < CIN / 4; ++c) {
            v2f A0 = *(const v2f*)(pa0 + 4 * c);        // A 16x4 fragments
            v2f A1 = *(const v2f*)(pa1 + 4 * c);
            const int rb = ((2 * c + hi) << 6) + lo;    // float2 elem index
#pragma unroll
            for (int t = 0; t < 4; ++t) {
                v2f B = *(const v2f*)&wlds[(rb + (t << 4)) * 2];
                accA[t] = __builtin_amdgcn_wmma_f32_16x16x4_f32(
                    false, A0, false, B, (short)0, accA[t], false, false);
                accB[t] = __builtin_amdgcn_wmma_f32_16x16x4_f32(
                    false, A1, false, B, (short)0, accB[t], false, false);
            }
        }
        __syncthreads();
    }

    // bias (folded GN constant) + SiLU
#pragma unroll
    for (int t = 0; t < 4; ++t) {
        float cb = Cb[(t << 4) + lo];
#pragma unroll
        for (int j = 0; j < 8; ++j) {
            accA[t][j] = silu_f(accA[t][j] + cb);
            accB[t][j] = silu_f(accB[t][j] + cb);
        }
    }

    if (SKIP) {
        // out += x @ Wskip + bskip, chained through WMMA with C = silu'd acc
        const float4* src = (const float4*)WpSkip;
#pragma unroll
        for (int i = tid; i < 32 * 16; i += 256) ((float4*)wlds)[i] = src[i];
        __syncthreads();
        const float* ps0 = skipAct + (size_t)vm0 * 32 + 2 * hi;
        const float* ps1 = skipAct + (size_t)vm1 * 32 + 2 * hi;
#pragma unroll
        for (int c = 0; c < 8; ++c) {
            v2f A0 = *(const v2f*)(ps0 + 4 * c);
            v2f A1 = *(const v2f*)(ps1 + 4 * c);
            const int rb = ((2 * c + hi) << 6) + lo;
#pragma unroll
            for (int t = 0; t < 4; ++t) {
                v2f B = *(const v2f*)&wlds[(rb + (t << 4)) * 2];
                accA[t] = __builtin_amdgcn_wmma_f32_16x16x4_f32(
                    false, A0, false, B, (short)0, accA[t], false, false);
                accB[t] = __builtin_amdgcn_wmma_f32_16x16x4_f32(
                    false, A1, false, B, (short)0, accB[t], false, false);
            }
        }
#pragma unroll
        for (int t = 0; t < 4; ++t) {
            float bs = bskip[(t << 4) + lo];
#pragma unroll
            for (int j = 0; j < 8; ++j) { accA[t][j] += bs; accB[t][j] += bs; }
        }
    }

    // D layout: lane holds column n = 16t+lo for rows M = j + 8*hi
#pragma unroll
    for (int j = 0; j < 8; ++j) {
        int v0 = m0 + j + (hi << 3);
        int v1 = v0 + 16;
        if (v0 < N) {
            float* po = out + (size_t)v0 * 64 + lo;
#pragma unroll
            for (int t = 0; t < 4; ++t) po[t << 4] = accA[t][j];
        }
        if (v1 < N) {
            float* po = out + (size_t)v1 * 64 + lo;
#pragma unroll
            for (int t = 0; t < 4; ++t) po[t << 4] = accB[t][j];
        }
    }
}

// --------- optional tail: feat_depth (tuple 2nd output) ---------------------
__global__ void write_tail(float* __restrict__ out, const int* __restrict__ fd,
                           size_t start, size_t total) {
    float v = (float)fd[0];
    for (size_t i = start + blockIdx.x * blockDim.x + threadIdx.x; i < total;
         i += (size_t)gridDim.x * blockDim.x)
        out[i] = v;
}

extern "C" void kernel_launch(void* const* d_in, const int* in_sizes, int n_in,
                              void* d_out, int out_size, void* d_ws, size_t ws_size,
                              hipStream_t stream) {
    const float* x      = (const float*)d_in[0];
    const int*   nbr    = (const int*)  d_in[1];
    const float* gamma1 = (const float*)d_in[2];
    const float* beta1  = (const float*)d_in[3];
    const float* W1     = (const float*)d_in[4];
    const float* gamma2 = (const float*)d_in[5];
    const float* beta2  = (const float*)d_in[6];
    const float* W2     = (const float*)d_in[7];
    const float* Wskip  = (const float*)d_in[8];
    const float* bskip  = (const float*)d_in[9];
    const int*   feat   = (const int*)  d_in[10];

    const int N = in_sizes[0] / 32;
    float* ws = (float*)d_ws;

    // workspace layout (floats); every word written before read each call
    size_t oH1  = 0;                          // h1: [N,64]
    size_t oP1  = oH1  + (size_t)N * 64;      // stage-1 partials (conv1 GN)
    size_t oP2  = oP1  + (size_t)PB * 16;     // stage-1 partials (conv2 GN)
    size_t oAB1 = oP2  + (size_t)PB * 16;     // a1[32], b1[32]
    size_t oAB2 = oAB1 + 64;                  // a2[64], b2[64]
    size_t oC1  = oAB2 + 128;                 // C1[64]
    size_t oC2  = oC1  + 64;                  // C2[64]
    size_t oWp1 = oC2  + 64;                  // packed a1*W1: 27*32*64
    size_t oWsk = oWp1 + (size_t)KTAPS * 32 * 64; // packed Wskip: 32*64
    size_t oWp2 = oWsk + (size_t)32 * 64;     // packed a2*W2: 27*64*64

    float* h1 = ws + oH1;
    const int nPairs = (N + 31) / 32;         // 32-voxel wave tiles
    const int gridC  = (nPairs + 7) / 8;

    // ---- GN1 stats, fold into W1 ----
    gn_partials<32><<<PB, 256, 0, stream>>>(x, N, ws + oP1);
    gn_finalize<32><<<1, 256, 0, stream>>>(ws + oP1, 1.0f / ((float)N * 4.0f),
                                           gamma1, beta1, W1, ws + oAB1, ws + oC1);
    prep_w<32><<<KTAPS, 256, 0, stream>>>(W1, ws + oAB1, ws + oWp1);
    prep_w<32><<<1, 256, 0, stream>>>(Wskip, nullptr, ws + oWsk);

    // ---- conv1 + SiLU -> h1 ----
    sparse_conv_wmma<32, false><<<gridC, 256, 0, stream>>>(
        x, nbr, ws + oWp1, ws + oC1, nullptr, nullptr, nullptr, h1, N, nPairs);

    // ---- GN2 stats, fold into W2 ----
    gn_partials<64><<<PB, 256, 0, stream>>>(h1, N, ws + oP2);
    gn_finalize<64><<<1, 256, 0, stream>>>(ws + oP2, 1.0f / ((float)N * 8.0f),
                                           gamma2, beta2, W2, ws + oAB2, ws + oC2);
    prep_w<64><<<KTAPS, 256, 0, stream>>>(W2, ws + oAB2, ws + oWp2);

    // ---- conv2 + SiLU + skip -> out ----
    sparse_conv_wmma<64, true><<<gridC, 256, 0, stream>>>(
        h1, nbr, ws + oWp2, ws + oC2, x, ws + oWsk, bskip,
        (float*)d_out, N, nPairs);

    // ---- tuple tail: feat_depth ----
    size_t main_out = (size_t)N * 64;
    if ((size_t)out_size > main_out)
        write_tail<<<1, 256, 0, stream>>>((float*)d_out, feat, main_out, (size_t)out_size);
}